// HGNN_conv_2508260901592
// MI455X (gfx1250) — compile-verified
//
#include <hip/hip_runtime.h>

typedef __attribute__((ext_vector_type(2))) float v2f;
typedef __attribute__((ext_vector_type(8))) float v8f;

#define FT 256            // IN_FT == OUT_FT == 256
#define N_NODES 100000    // divisible by 16 -> 6250 M-tiles

// ---------------------------------------------------------------------------
// Zero the output (SpMM accumulates with atomics).
// ---------------------------------------------------------------------------
__global__ void hgnn_zero(float* __restrict__ out, long n) {
    long i = (long)blockIdx.x * blockDim.x + threadIdx.x;
    long stride = (long)gridDim.x * blockDim.x;
    for (; i < n; i += stride) out[i] = 0.0f;
}

// ---------------------------------------------------------------------------
// One-shot 256x256 transpose of W into workspace so the WMMA B-fragment
// becomes a contiguous per-lane float2 load (same b64 pattern as A).
// ---------------------------------------------------------------------------
__global__ void hgnn_transpose_w(const float* __restrict__ w, float* __restrict__ wt) {
    const int k = blockIdx.x;      // 0..255
    const int n = threadIdx.x;     // 0..255
    wt[(long)n * FT + k] = w[(long)k * FT + n];
}

// ---------------------------------------------------------------------------
// y = x @ W + b  via V_WMMA_F32_16X16X4_F32.
// One wave computes a 16x64 strip of y: 4 accumulator tiles sharing each
// A fragment (4 WMMAs per A load). 6250 M-tiles x 4 N-quarters = 25000 waves
// = 3125 blocks of 8 waves, exactly filled -> EXEC all ones (WMMA requires).
//
// Per-lane fragment layout (ISA 7.12.2, 32-bit operands, wave32):
//   A (16x4):  m = lane%16, holds K = 2*(lane/16) .. +1  -> contiguous f2
//   B (4x16):  n = lane%16, holds K = 2*(lane/16) .. +1  -> contiguous f2
//              from W^T (wt[n][k])
//   C/D (16x16, 8 VGPR): n = lane%16, M = 8*(lane/16) + v
// ---------------------------------------------------------------------------
__global__ __launch_bounds__(256) void hgnn_gemm_bias(
    const float* __restrict__ x, const float* __restrict__ wt,
    const float* __restrict__ bias, float* __restrict__ y) {

    const int lane = threadIdx.x & 31;
    const int wave = threadIdx.x >> 5;
    const int wid  = blockIdx.x * 8 + wave;      // 0 .. 24999
    const int tm   = (wid >> 2) << 4;            // row-tile base (16 rows)
    const int tn   = (wid & 3) << 6;             // col base (64 cols)

    const int mn    = lane & 15;                 // m for A, n for B/C/D
    const int khalf = (lane >> 4) << 1;          // 0 or 2

    const float* xrow = x  + (long)(tm + mn) * FT + khalf;
    const float* b0   = wt + (long)(tn + mn)      * FT + khalf;
    const float* b1   = wt + (long)(tn + 16 + mn) * FT + khalf;
    const float* b2   = wt + (long)(tn + 32 + mn) * FT + khalf;
    const float* b3   = wt + (long)(tn + 48 + mn) * FT + khalf;

    v8f c0 = {}, c1 = {}, c2 = {}, c3 = {};
    #pragma unroll 4
    for (int k = 0; k < FT; k += 4) {
        v2f a  = *(const v2f*)(xrow + k);
        v2f w0 = *(const v2f*)(b0 + k);
        v2f w1 = *(const v2f*)(b1 + k);
        v2f w2 = *(const v2f*)(b2 + k);
        v2f w3 = *(const v2f*)(b3 + k);
        c0 = __builtin_amdgcn_wmma_f32_16x16x4_f32(false, a, false, w0, (short)0, c0, false, false);
        c1 = __builtin_amdgcn_wmma_f32_16x16x4_f32(false, a, false, w1, (short)0, c1, false, false);
        c2 = __builtin_amdgcn_wmma_f32_16x16x4_f32(false, a, false, w2, (short)0, c2, false, false);
        c3 = __builtin_amdgcn_wmma_f32_16x16x4_f32(false, a, false, w3, (short)0, c3, false, false);
    }

    // D layout: row m = tm + 8*(lane>>4) + v, col = tn + j*16 + mn
    float* yout = y + (long)(tm + ((lane >> 4) << 3)) * FT + tn + mn;
    const float bv0 = bias[tn + mn];
    const float bv1 = bias[tn + 16 + mn];
    const float bv2 = bias[tn + 32 + mn];
    const float bv3 = bias[tn + 48 + mn];
    #pragma unroll
    for (int v = 0; v < 8; ++v) {
        yout[(long)v * FT +  0] = c0[v] + bv0;
        yout[(long)v * FT + 16] = c1[v] + bv1;
        yout[(long)v * FT + 32] = c2[v] + bv2;
        yout[(long)v * FT + 48] = c3[v] + bv3;
    }
}

// ---------------------------------------------------------------------------
// out[r] += v * y[c]  for each (r, c, v) edge.
// 64 threads cooperate on one edge: thread t handles float4 #t of the
// 256-float feature row (float4 gather + 4 global_atomic_add_f32, which are
// non-returning -> STOREcnt, and land in the L2 where `out` fits).
// 256-thread block = 4 edges.
// ---------------------------------------------------------------------------
__global__ __launch_bounds__(256) void hgnn_spmm(
    const int* __restrict__ rows, const int* __restrict__ cols,
    const float* __restrict__ vals, const float* __restrict__ y,
    float* __restrict__ out, int nnz) {

    const int edge = blockIdx.x * 4 + (threadIdx.x >> 6);
    if (edge >= nnz) return;
    const int t = threadIdx.x & 63;              // float4 slot within the row

    const int   r = rows[edge];
    const int   c = cols[edge];
    const float v = vals[edge];

    const float4 g = ((const float4*)(y + (long)c * FT))[t];
    float* dst = out + (long)r * FT + t * 4;

    atomicAdd(dst + 0, v * g.x);
    atomicAdd(dst + 1, v * g.y);
    atomicAdd(dst + 2, v * g.z);
    atomicAdd(dst + 3, v * g.w);
}

// ---------------------------------------------------------------------------
extern "C" void kernel_launch(void* const* d_in, const int* in_sizes, int n_in,
                              void* d_out, int out_size, void* d_ws, size_t ws_size,
                              hipStream_t stream) {
    const float* x      = (const float*)d_in[0];   // [N_NODES, FT]
    const int*   G_rows = (const int*)  d_in[1];   // [NNZ]
    const int*   G_cols = (const int*)  d_in[2];   // [NNZ]
    const float* G_vals = (const float*)d_in[3];   // [NNZ]
    const float* weight = (const float*)d_in[4];   // [FT, FT]
    const float* bias   = (const float*)d_in[5];   // [FT]
    float*       out    = (float*)d_out;           // [N_NODES, FT]

    float* y  = (float*)d_ws;                      // N_NODES*FT floats
    float* wt = y + (long)N_NODES * FT;            // FT*FT floats (W^T)

    const int nnz = in_sizes[1];

    // 1) out = 0
    hgnn_zero<<<2048, 256, 0, stream>>>(out, (long)N_NODES * FT);

    // 2) W^T into workspace (256 KB, one-shot)
    hgnn_transpose_w<<<FT, FT, 0, stream>>>(weight, wt);

    // 3) y = x @ W + b   (one wave per 16x64 strip; 8 waves/block)
    const int waves = (N_NODES / 16) * (FT / 64);  // 25000
    hgnn_gemm_bias<<<waves / 8, 256, 0, stream>>>(x, wt, bias, y);

    // 4) out[r] += v * y[c]   (4 edges per block)
    hgnn_spmm<<<(nnz + 3) / 4, 256, 0, stream>>>(G_rows, G_cols, G_vals, y, out, nnz);
}